// SparseBMM_7284264534722
// MI455X (gfx1250) — compile-verified
//
#include <hip/hip_runtime.h>

typedef __attribute__((ext_vector_type(2))) float v2f;
typedef __attribute__((ext_vector_type(8))) float v8f;
typedef __attribute__((ext_vector_type(4))) float f4;
typedef __attribute__((ext_vector_type(4))) unsigned int u4;

#define TILE_M 128
#define TILE_K 128
#define LDA 132     // padded LDS stride for A tile (floats): 528B row shift -> conflict-free b64 frag loads
#define SWZP 288    // dwords per k-pair row of swizzled B tile: 1152B = 128 mod 256 -> lane halves on disjoint banks

// C[b] = A[b] @ B[b]; A is tile-sparse at 128x128 granularity -> skip zero tiles.
__launch_bounds__(256)
__global__ void sparse_bmm_f32_wmma(const float* __restrict__ A,
                                    const float* __restrict__ B,
                                    float* __restrict__ C,
                                    int M, int K, int N) {
  __shared__ __align__(16) float lds_a[TILE_M * LDA];          // 67,584 B
  __shared__ __align__(16) float lds_b[(TILE_K / 2) * SWZP];   // 73,728 B, k-pair interleaved
  __shared__ int s_wflag[8];

  const int tid  = threadIdx.x;
  const int lane = tid & 31;
  const int wave = tid >> 5;      // 8 waves (wave32)
  const int mt   = blockIdx.x;    // M tile index
  const int bb   = blockIdx.y;    // batch index

  const float* Ab = A + (size_t)bb * M * K + (size_t)mt * TILE_M * K;
  const float* Bb = B + (size_t)bb * K * N;
  float*       Cb = C + (size_t)bb * M * N + (size_t)mt * TILE_M * N;

  v8f acc[8] = {};                // zero accumulators (also correct output for all-zero rows)

  // cooperative tile load mapping: 32 float4 per 128-float row
  const int f  = tid & 31;        // float4 column within row
  const int r0 = tid >> 5;        // base row (8 rows per sweep)

  for (int kt = 0; kt < K; kt += TILE_K) {
    __syncthreads();              // all LDS reads of previous k-tile complete

    // ---- stream A tile (128x128 f32) global -> LDS, non-temporal, track nonzero bits ----
    unsigned int nzor = 0u;
    #pragma unroll
    for (int i = 0; i < 16; ++i) {
      const int row = r0 + i * 8;
      const f4* src = (const f4*)(Ab + (size_t)row * K + kt) + f;
      f4 v = __builtin_nontemporal_load(src);
      u4 u;
      __builtin_memcpy(&u, &v, sizeof(u));
      nzor |= u.x | u.y | u.z | u.w;
      *(f4*)&lds_a[row * LDA + f * 4] = v;
    }
    // mask sign bits: zeroed tiles hold -0.0 (a * 0.0f), which must count as zero
    unsigned long long any = __ballot((nzor & 0x7FFFFFFFu) != 0u);
    if (lane == 0) s_wflag[wave] = (any != 0ull) ? 1 : 0;
    __syncthreads();              // A tile + flags visible

    const int nz = s_wflag[0] | s_wflag[1] | s_wflag[2] | s_wflag[3]
                 | s_wflag[4] | s_wflag[5] | s_wflag[6] | s_wflag[7];
    if (!nz) continue;            // whole 128x128 A tile is zero: skip B load + all WMMAs

    // ---- load B tile (128 x N=128), swizzle to k-pair interleaved LDS layout ----
    // LDS dword index for element B[2p+q][n] (local k = 2p+q) is  p*SWZP + n*2 + q,
    // so a WMMA B fragment (two k's, same n) is one aligned b64.
    #pragma unroll
    for (int i = 0; i < 8; ++i) {
      const int pair = wave + i * 8;            // k-pair index 0..63
      const int krow = kt + pair * 2;
      f4 v0 = *((const f4*)(Bb + (size_t)(krow + 0) * N) + f);
      f4 v1 = *((const f4*)(Bb + (size_t)(krow + 1) * N) + f);
      f4 out0 = {v0.x, v1.x, v0.y, v1.y};
      f4 out1 = {v0.z, v1.z, v0.w, v1.w};
      float* dst = &lds_b[pair * SWZP + f * 8];
      *(f4*)(dst + 0) = out0;
      *(f4*)(dst + 4) = out1;
    }
    __syncthreads();              // B tile visible

    // ---- WMMA: wave owns M rows [16*wave, 16*wave+16), all 8 N sub-tiles ----
    // A frag (16x4 f32): lanes 0-15 -> K={0,1}, lanes 16-31 -> K={2,3}; one b64 per lane.
    // B frag: one b64 per lane from the swizzled tile (k-pair p, column n).
    const int mrow  = 16 * wave + (lane & 15);
    const int khalf = lane >> 4;                // 0 or 1 (k-pair offset)
    const int ncol  = lane & 15;
    #pragma unroll 4
    for (int kk = 0; kk < TILE_K; kk += 4) {
      const int p = (kk >> 1) + khalf;
      v2f afrag = *(const v2f*)&lds_a[mrow * LDA + kk + 2 * khalf];
      #pragma unroll
      for (int n = 0; n < 8; ++n) {
        v2f bfrag = *(const v2f*)&lds_b[p * SWZP + (n * 16 + ncol) * 2];
        acc[n] = __builtin_amdgcn_wmma_f32_16x16x4_f32(
            false, afrag, false, bfrag, (short)0, acc[n], false, false);
      }
    }
  }

  // ---- store C tile: C/D layout VGPR j -> M=j (lanes 0-15) / M=j+8 (lanes 16-31), lane -> N ----
  {
    const int mbase = 16 * wave + 8 * (lane >> 4);
    const int ncol  = lane & 15;
    #pragma unroll
    for (int n = 0; n < 8; ++n) {
      #pragma unroll
      for (int j = 0; j < 8; ++j) {
        __builtin_nontemporal_store(acc[n][j],
            &Cb[(size_t)(mbase + j) * N + n * 16 + ncol]);
      }
    }
  }
}

extern "C" void kernel_launch(void* const* d_in, const int* in_sizes, int n_in,
                              void* d_out, int out_size, void* d_ws, size_t ws_size,
                              hipStream_t stream) {
  const float* A = (const float*)d_in[0];   // [8, 4096, 4096] f32, tile-sparse
  const float* B = (const float*)d_in[1];   // [8, 4096, 128]  f32
  float*       C = (float*)d_out;           // [8, 4096, 128]  f32

  const int Bt = 8, M = 4096, K = 4096, N = 128;
  (void)in_sizes; (void)n_in; (void)out_size; (void)d_ws; (void)ws_size;

  dim3 grid(M / TILE_M, Bt);   // 32 x 8 = 256 workgroups
  sparse_bmm_f32_wmma<<<grid, 256, 0, stream>>>(A, B, C, M, K, N);
}